// MultiHeadAttention_50964081934816
// MI455X (gfx1250) — compile-verified
//
#include <hip/hip_runtime.h>
#include <hip/hip_bf16.h>

// MHA forward for B=2, S=2048, H=16, D=64 (fp32), returning (out, attn).
// All GEMMs run on V_WMMA_F32_16X16X4_F32 (CDNA5 fp32 WMMA, wave32).
// Memory-bound on the 537MB attn tensor -> fp32 WMMA is the right precision.

#define B_  2
#define S_  2048
#define H_  16
#define D_  64
#define HD_ 1024           // H*D
#define M_  (B_ * S_)      // 4096 flattened rows
#define BH_ (B_ * H_)      // 32

typedef __attribute__((ext_vector_type(2))) float v2f;
typedef __attribute__((ext_vector_type(8))) float v8f;

// D = A(16x4) * B(4x16) + C ; fp32, wave32.
// 8-arg form: (neg_a, A, neg_b, B, c_mod, C, reuse_a, reuse_b)
__device__ __forceinline__ v8f wmma4(v2f a, v2f b, v8f c) {
    return __builtin_amdgcn_wmma_f32_16x16x4_f32(
        false, a, false, b, (short)0, c, false, false);
}

// ---------------------------------------------------------------------------
// Kernel 1: fused QKV projection.
// p=0: Q = query@Wq+bq ; p=1: K = value@Wk+bk ; p=2: V = value@Wv+bv
// GEMM M=4096, K=64, N=1024; output written in [B,H,S,D] layout.
// One 16x16 tile per wave; 3*256*64 = 49152 tiles; 8 waves/block -> 6144 blocks.
// ---------------------------------------------------------------------------
__global__ void __launch_bounds__(256)
qkv_proj_kernel(const float* __restrict__ query, const float* __restrict__ value,
                const float* __restrict__ Wq, const float* __restrict__ bq,
                const float* __restrict__ Wk, const float* __restrict__ bk,
                const float* __restrict__ Wv, const float* __restrict__ bv,
                float* __restrict__ Qb, float* __restrict__ Kb, float* __restrict__ Vb)
{
    const int wave = threadIdx.x >> 5;
    const int lane = threadIdx.x & 31;
    const int half = lane >> 4;    // K-pair select (A layout)
    const int idx  = lane & 15;    // M row (A) / N col (B)

    int tile = blockIdx.x * 8 + wave;
    const int p   = tile / (256 * 64);
    const int rem = tile % (256 * 64);
    const int mt  = rem / 64;
    const int nt  = rem % 64;

    const float* A    = (p == 0) ? query : value;
    const float* W    = (p == 0) ? Wq : (p == 1) ? Wk : Wv;
    const float* bias = (p == 0) ? bq : (p == 1) ? bk : bv;
    float*       out  = (p == 0) ? Qb : (p == 1) ? Kb : Vb;

    const int m0 = mt * 16, n0 = nt * 16;
    // A frag: row = m0+idx, K elements (kb+2*half, kb+2*half+1) contiguous.
    const float* Arow = A + (m0 + idx) * D_ + 2 * half;
    // B frag as B^T(NxK): col = n0+idx; W is [64,1024] row-major, stride HD_.
    const float* Bcol = W + (size_t)(2 * half) * HD_ + (n0 + idx);

    v8f c = {};
#pragma unroll
    for (int kb = 0; kb < D_; kb += 4) {
        v2f a = *(const v2f*)(Arow + kb);
        v2f b;
        b.x = Bcol[(size_t)kb * HD_];
        b.y = Bcol[(size_t)(kb + 1) * HD_];
        c = wmma4(a, b, c);
    }

    const float bn = bias[n0 + idx];
#pragma unroll
    for (int r = 0; r < 8; ++r) {
        const int gm = m0 + r + 8 * half;      // row in [0, B*S)
        const int gn = n0 + idx;               // col in [0, H*D)
        const int bb = gm >> 11;               // / S_
        const int ss = gm & (S_ - 1);
        const int hh = gn >> 6;                // / D_
        const int dd = gn & 63;
        out[(((size_t)(bb * H_ + hh)) * S_ + ss) * D_ + dd] = c[r] + bn;
    }
}

// ---------------------------------------------------------------------------
// Kernel 2: scores = (Q @ K^T) * 0.125, masked, written into attn buffer.
// Per (b,h): M=N=2048, K=64. Each wave computes a 16x64 strip (4 accumulators,
// A-frag reused 4x). 32 * 128 * 32 = 131072 waves -> 16384 blocks.
// ---------------------------------------------------------------------------
__global__ void __launch_bounds__(256)
scores_kernel(const float* __restrict__ Qb, const float* __restrict__ Kb,
              const int* __restrict__ mask, float* __restrict__ attn)
{
    const int wave = threadIdx.x >> 5;
    const int lane = threadIdx.x & 31;
    const int half = lane >> 4;
    const int idx  = lane & 15;

    int t = blockIdx.x * 8 + wave;
    const int nq = t & 31;  t >>= 5;    // 64-wide N strip index
    const int mt = t & 127; t >>= 7;
    const int bh = t;                    // [0, 32)
    const int bb = bh >> 4;              // / H_

    const float* Q = Qb + (size_t)bh * S_ * D_;
    const float* K = Kb + (size_t)bh * S_ * D_;
    const int m0 = mt * 16;

    const float* Arow = Q + (m0 + idx) * D_ + 2 * half;
    v8f acc[4] = {};
#pragma unroll
    for (int kb = 0; kb < D_; kb += 4) {
        v2f a = *(const v2f*)(Arow + kb);
        // B = K^T (DxS); B^T = K itself (SxD): lane idx is score-col n.
        const float* Kbase = K + kb + 2 * half;
#pragma unroll
        for (int j = 0; j < 4; ++j) {
            v2f b = *(const v2f*)(Kbase + (size_t)(nq * 64 + j * 16 + idx) * D_);
            acc[j] = wmma4(a, b, acc[j]);
        }
    }

    const float scale = 0.125f;          // 1/sqrt(64)
    const size_t attnBase = (size_t)bh * S_ * S_;
    const int* mrow = mask + (size_t)bb * S_ * S_;
#pragma unroll
    for (int j = 0; j < 4; ++j) {
        const int kcol = nq * 64 + j * 16 + idx;
#pragma unroll
        for (int r = 0; r < 8; ++r) {
            const int q = m0 + r + 8 * half;
            float v = acc[j][r] * scale;
            if (mrow[(size_t)q * S_ + kcol] == 0) v = -1e9f;
            attn[attnBase + (size_t)q * S_ + kcol] = v;
        }
    }
}

// ---------------------------------------------------------------------------
// Kernel 3: row softmax in-place over attn. One 256-thread block per row
// (B*H*S = 65536 rows), 8 elems/thread kept in registers: 1 read + 1 write.
// ---------------------------------------------------------------------------
__global__ void __launch_bounds__(256)
softmax_kernel(float* __restrict__ attn)
{
    const size_t row = blockIdx.x;
    float* p = attn + row * S_;
    const int tid = threadIdx.x;

    float x[8];
    float m = -1e30f;
#pragma unroll
    for (int i = 0; i < 8; ++i) {
        x[i] = p[tid + i * 256];
        m = fmaxf(m, x[i]);
    }
    __shared__ float red[256];
    red[tid] = m;
    __syncthreads();
    for (int s = 128; s > 0; s >>= 1) {
        if (tid < s) red[tid] = fmaxf(red[tid], red[tid + s]);
        __syncthreads();
    }
    m = red[0];
    __syncthreads();

    float sum = 0.0f;
#pragma unroll
    for (int i = 0; i < 8; ++i) {
        x[i] = __expf(x[i] - m);
        sum += x[i];
    }
    red[tid] = sum;
    __syncthreads();
    for (int s = 128; s > 0; s >>= 1) {
        if (tid < s) red[tid] += red[tid + s];
        __syncthreads();
    }
    const float inv = 1.0f / red[0];
#pragma unroll
    for (int i = 0; i < 8; ++i) p[tid + i * 256] = x[i] * inv;
}

// ---------------------------------------------------------------------------
// Kernel 4: ctx = attn @ V. Per (b,h): M=2048, K=2048, N=64.
// Each wave: 16 rows x full 64 cols (4 accumulators, A-frag reuse 4x),
// K accumulation over 2048 in steps of 4. 32*128 = 4096 waves -> 512 blocks.
// ctx stored in [B,S,H*D] layout for the output projection.
// ---------------------------------------------------------------------------
__global__ void __launch_bounds__(256)
ctx_kernel(const float* __restrict__ attn, const float* __restrict__ Vb,
           float* __restrict__ ctx)
{
    const int wave = threadIdx.x >> 5;
    const int lane = threadIdx.x & 31;
    const int half = lane >> 4;
    const int idx  = lane & 15;

    int t = blockIdx.x * 8 + wave;
    const int mt = t & 127;
    const int bh = t >> 7;               // [0, 32)
    const int bb = bh >> 4;
    const int hh = bh & 15;

    const float* A = attn + ((size_t)bh * S_ + mt * 16 + idx) * S_ + 2 * half;
    const float* V = Vb + (size_t)bh * S_ * D_;

    v8f acc[4] = {};
#pragma unroll 4
    for (int kb = 0; kb < S_; kb += 4) {
        v2f a = *(const v2f*)(A + kb);
        const float* Vk0 = V + (size_t)(kb + 2 * half) * D_;   // row K
        const float* Vk1 = Vk0 + D_;                            // row K+1
#pragma unroll
        for (int j = 0; j < 4; ++j) {
            v2f b;
            b.x = Vk0[j * 16 + idx];
            b.y = Vk1[j * 16 + idx];
            acc[j] = wmma4(a, b, acc[j]);
        }
    }

#pragma unroll
    for (int j = 0; j < 4; ++j) {
#pragma unroll
        for (int r = 0; r < 8; ++r) {
            const int q = mt * 16 + r + 8 * half;
            ctx[((size_t)bb * S_ + q) * HD_ + hh * 64 + j * 16 + idx] = acc[j][r];
        }
    }
}

// ---------------------------------------------------------------------------
// Kernel 5: out = ctx @ Wo + bo. M=4096, K=1024, N=64.
// 256 * 4 = 1024 tiles -> 128 blocks.
// ---------------------------------------------------------------------------
__global__ void __launch_bounds__(256)
outproj_kernel(const float* __restrict__ ctx, const float* __restrict__ Wo,
               const float* __restrict__ bo, float* __restrict__ out)
{
    const int wave = threadIdx.x >> 5;
    const int lane = threadIdx.x & 31;
    const int half = lane >> 4;
    const int idx  = lane & 15;

    int t = blockIdx.x * 8 + wave;
    const int nt = t & 3;
    const int mt = t >> 2;               // [0, 256)
    const int m0 = mt * 16, n0 = nt * 16;

    const float* Arow = ctx + (size_t)(m0 + idx) * HD_ + 2 * half;
    const float* Bcol = Wo + (size_t)(2 * half) * D_ + (n0 + idx);

    v8f c = {};
#pragma unroll 8
    for (int kb = 0; kb < HD_; kb += 4) {
        v2f a = *(const v2f*)(Arow + kb);
        v2f b;
        b.x = Bcol[(size_t)kb * D_];
        b.y = Bcol[(size_t)(kb + 1) * D_];
        c = wmma4(a, b, c);
    }

    const float bn = bo[n0 + idx];
#pragma unroll
    for (int r = 0; r < 8; ++r) {
        const int gm = m0 + r + 8 * half;
        out[(size_t)gm * D_ + n0 + idx] = c[r] + bn;
    }
}

// ---------------------------------------------------------------------------
extern "C" void kernel_launch(void* const* d_in, const int* in_sizes, int n_in,
                              void* d_out, int out_size, void* d_ws, size_t ws_size,
                              hipStream_t stream) {
    const float* query = (const float*)d_in[0];
    const float* value = (const float*)d_in[1];
    const int*   mask  = (const int*)d_in[2];
    const float* Wq = (const float*)d_in[3];
    const float* bq = (const float*)d_in[4];
    const float* Wk = (const float*)d_in[5];
    const float* bk = (const float*)d_in[6];
    const float* Wv = (const float*)d_in[7];
    const float* bv = (const float*)d_in[8];
    const float* Wo = (const float*)d_in[9];
    const float* bo = (const float*)d_in[10];

    float* out  = (float*)d_out;                        // [B,S,D]
    float* attn = out + (size_t)B_ * S_ * D_;           // [B,H,S,S]

    const size_t QKV = (size_t)B_ * H_ * S_ * D_;       // 4,194,304 floats each
    float* Qb  = (float*)d_ws;
    float* Kb  = Qb + QKV;
    float* Vb  = Kb + QKV;
    float* ctx = Vb + QKV;                              // [B,S,H*D], 16MB

    qkv_proj_kernel<<<6144, 256, 0, stream>>>(query, value, Wq, bq, Wk, bk, Wv, bv,
                                              Qb, Kb, Vb);
    scores_kernel<<<16384, 256, 0, stream>>>(Qb, Kb, mask, attn);
    softmax_kernel<<<(unsigned)((size_t)B_ * H_ * S_), 256, 0, stream>>>(attn);
    ctx_kernel<<<512, 256, 0, stream>>>(attn, Vb, ctx);
    outproj_kernel<<<128, 256, 0, stream>>>(ctx, Wo, bo, out);
}